// FCOSOutputs_51496657879410
// MI455X (gfx1250) — compile-verified
//
#include <hip/hip_runtime.h>
#include <hip/hip_bf16.h>
#include <stdint.h>

typedef __attribute__((ext_vector_type(2))) float    v2f;
typedef __attribute__((ext_vector_type(8))) float    v8f;
typedef __attribute__((ext_vector_type(4))) uint32_t u32x4;
typedef __attribute__((ext_vector_type(8))) int      i32x8;
typedef __attribute__((ext_vector_type(4))) int      i32x4;

#define FCOS_L      20267
#define FCOS_B      8
#define FCOS_G      64
#define FCOS_INF    100000000.0f
#define FCOS_NCLS   80
#define FCOS_RADIUS 1.5f

// Level boundaries (cumulative location counts): 15200, 19000, 19950, 20197, 20267
__device__ __forceinline__ void level_of(int idx, float& st, float& lo, float& hi) {
  st = 8.0f;  lo = -1.0f;  hi = 64.0f;
  if (idx >= 15200) { st = 16.0f;  lo = 64.0f;  hi = 128.0f; }
  if (idx >= 19000) { st = 32.0f;  lo = 128.0f; hi = 256.0f; }
  if (idx >= 19950) { st = 64.0f;  lo = 256.0f; hi = 512.0f; }
  if (idx >= 20197) { st = 128.0f; lo = 512.0f; hi = FCOS_INF; }
}

// Stage 256 contiguous f32 (64 boxes) from global into LDS via the Tensor Data
// Mover. D# per CDNA5 ISA §8.3/8.4: 2D tensor, data_size=4B, tile 256x1.
__device__ __forceinline__ void tdm_load_boxes(const float* __restrict__ src,
                                               float* lds_dst) {
  const uint64_t ga   = (uint64_t)(uintptr_t)src;
  const uint32_t ldsa = (uint32_t)(uintptr_t)lds_dst;  // generic LDS addr[31:0] == LDS offset

  u32x4 g0;
  g0[0] = 1u;                                           // count=1 (valid user D#)
  g0[1] = ldsa;                                         // lds_addr
  g0[2] = (uint32_t)(ga & 0xffffffffu);                 // global_addr[31:0]
  g0[3] = (uint32_t)((ga >> 32) & 0x01ffffffu)          // global_addr[56:32]
        | (2u << 30);                                   // type=2 ("image")

  i32x8 g1;
  g1[0] = (int)(2u << 16);     // workgroup_mask=0, data_size=2 (4 bytes)
  g1[1] = (int)(256u << 16);   // tensor_dim0 = 256  (bits 79:48)
  g1[2] = (int)(1u << 16);     // tensor_dim1 = 1    (bits 111:80)
  g1[3] = (int)(256u << 16);   // tile_dim0   = 256  (bits 127:112)
  g1[4] = 1;                   // tile_dim1   = 1    (bits 143:128), tile_dim2=0
  g1[5] = 256;                 // tensor_dim0_stride = 256 (bits 207:160)
  g1[6] = (int)(256u << 16);   // tensor_dim1_stride = 256 (bits 255:208)
  g1[7] = 0;

  i32x4 gz = {0, 0, 0, 0};
#if defined(__clang_major__) && (__clang_major__ >= 23)
  i32x8 gz8 = {0, 0, 0, 0, 0, 0, 0, 0};
  __builtin_amdgcn_tensor_load_to_lds(g0, g1, gz, gz, gz8, 0);
#else
  __builtin_amdgcn_tensor_load_to_lds(g0, g1, gz, gz, 0);
#endif
  __builtin_amdgcn_s_wait_tensorcnt((short)0);
}

__global__ __launch_bounds__(256)
void fcos_assign_wmma(const float* __restrict__ locations,   // [L,2]
                      const float* __restrict__ gt_boxes,    // [B,64,4]
                      const int*   __restrict__ gt_classes,  // [B,64]
                      float* __restrict__ out)               // labels BL | reg 4BL | ctr BL
{
  __shared__ __align__(16) float s_box[FCOS_G * 4];
  __shared__ int s_cls[FCOS_G];

  const int b    = blockIdx.y;
  const int tid  = threadIdx.x;
  const int wave = tid >> 5;
  const int lane = tid & 31;
  const int hlf  = lane >> 4;   // 0: rows 0..7 / K0,K1 ; 1: rows 8..15 / K2,K3
  const int n16  = lane & 15;   // box column within tile

  __builtin_prefetch(locations, 0, 0);

  if (wave == 0) tdm_load_boxes(gt_boxes + b * FCOS_G * 4, s_box);  // async DMA -> LDS
  if (tid < FCOS_G) s_cls[tid] = gt_classes[b * FCOS_G + tid];
  __syncthreads();

  const int base = (blockIdx.x * 8 + wave) * 16;   // first location of this wave's tile
  if (base >= FCOS_L) return;                      // wave-uniform

  // Per-lane location attributes for location (base + n16), clamped; duplicated in both halves.
  int liA = base + n16; if (liA >= FCOS_L) liA = FCOS_L - 1;
  const float lx = locations[liA * 2 + 0];
  const float ly = locations[liA * 2 + 1];
  float stA, loA, hiA;
  level_of(liA, stA, loA, hiA);
  const float srA = stA * FCOS_RADIUS;

  // A fragment (16x4 f32): row m = [x_m, y_m, sr_m, 1]
  v2f A;
  A.x = hlf ? srA  : lx;   // K2 : K0
  A.y = hlf ? 1.0f : ly;   // K3 : K1

  // Row-wise (M) level constants for this half's rows m = r + 8*hlf
  float loM[8], hiM[8];
#pragma unroll
  for (int r = 0; r < 8; ++r) {
    int mi = base + r + 8 * hlf; if (mi >= FCOS_L) mi = FCOS_L - 1;
    float s_, l_, h_;
    level_of(mi, s_, l_, h_);
    loM[r] = l_; hiM[r] = h_;
  }

  float bestA[8];
  int   bestI[8];
#pragma unroll
  for (int r = 0; r < 8; ++r) { bestA[r] = 3.0e38f; bestI[r] = 1 << 30; }

  const float fh = hlf ? 1.0f : 0.0f;

  for (int bt = 0; bt < 4; ++bt) {
    const int   g    = bt * 16 + n16;
    const float bx0  = s_box[g * 4 + 0];
    const float by0  = s_box[g * 4 + 1];
    const float bx1  = s_box[g * 4 + 2];
    const float by1  = s_box[g * 4 + 3];
    const float cx   = (bx0 + bx1) * 0.5f;
    const float cy   = (by0 + by1) * 0.5f;
    const float area = (bx1 - bx0) * (by1 - by0);

    // B fragments (4x16 f32), column j = [c_x, c_y, c_sr, const_j]:
    //   v0: half0 -> K0 (c_x), half1 -> K2 (c_sr); v1: half0 -> K1 (c_y), half1 -> K3 (const)
    v2f Bl, Bt, Br, Bb, Bm1, Bm2, Bm3, Bm4;
    Bl.x  = hlf ? 0.0f : 1.0f;   Bl.y  = hlf ? -bx0 : 0.0f;   // l  = x - bx0
    Bt.x  = 0.0f;                Bt.y  = hlf ? -by0 : 1.0f;   // t  = y - by0
    Br.x  = hlf ? 0.0f : -1.0f;  Br.y  = hlf ?  bx1 : 0.0f;   // r  = bx1 - x
    Bb.x  = 0.0f;                Bb.y  = hlf ?  by1 : -1.0f;  // b  = by1 - y
    Bm1.x = 1.0f;                Bm1.y = hlf ? -cx  : 0.0f;   // x + sr - cx
    Bm2.x = hlf ? 1.0f : -1.0f;  Bm2.y = hlf ?  cx  : 0.0f;   // cx + sr - x
    Bm3.x = fh;                  Bm3.y = hlf ? -cy  : 1.0f;   // y + sr - cy
    Bm4.x = fh;                  Bm4.y = hlf ?  cy  : -1.0f;  // cy + sr - y

    v8f z = {};
    v8f Dl  = __builtin_amdgcn_wmma_f32_16x16x4_f32(false, A, false, Bl,  (short)0, z, false, false);
    v8f Dt  = __builtin_amdgcn_wmma_f32_16x16x4_f32(false, A, false, Bt,  (short)0, z, false, false);
    v8f Dr  = __builtin_amdgcn_wmma_f32_16x16x4_f32(false, A, false, Br,  (short)0, z, false, false);
    v8f Db  = __builtin_amdgcn_wmma_f32_16x16x4_f32(false, A, false, Bb,  (short)0, z, false, false);
    v8f Dm1 = __builtin_amdgcn_wmma_f32_16x16x4_f32(false, A, false, Bm1, (short)0, z, false, false);
    v8f Dm2 = __builtin_amdgcn_wmma_f32_16x16x4_f32(false, A, false, Bm2, (short)0, z, false, false);
    v8f Dm3 = __builtin_amdgcn_wmma_f32_16x16x4_f32(false, A, false, Bm3, (short)0, z, false, false);
    v8f Dm4 = __builtin_amdgcn_wmma_f32_16x16x4_f32(false, A, false, Bm4, (short)0, z, false, false);

#pragma unroll
    for (int r = 0; r < 8; ++r) {
      const float l  = Dl[r], t = Dt[r], rr = Dr[r], bb = Db[r];
      const float mx = fmaxf(fmaxf(l, t), fmaxf(rr, bb));
      const float ins = fminf(fminf(fminf(Dm1[r], l), fminf(Dm2[r], rr)),
                              fminf(fminf(Dm3[r], t), fminf(Dm4[r], bb)));
      // branchless: bitwise AND/OR of comparisons (no exec-mask short circuits)
      const int ok = (int)(ins > 0.0f) & (int)(mx >= loM[r]) & (int)(mx <= hiM[r]);
      const float cand = ok ? area : FCOS_INF;
      const int better = (int)(cand < bestA[r]) |
                         ((int)(cand == bestA[r]) & (int)(g < bestI[r]));
      bestA[r] = better ? cand : bestA[r];
      bestI[r] = better ? g    : bestI[r];
    }
  }

  // Argmin over the 16 N-lanes of each half (first-occurrence tie-break like jnp.argmin)
#pragma unroll
  for (int r = 0; r < 8; ++r) {
    float a = bestA[r]; int i = bestI[r];
#pragma unroll
    for (int off = 1; off < 16; off <<= 1) {
      const float oa = __shfl_xor(a, off, 32);
      const int   oi = __shfl_xor(i, off, 32);
      const int take = (int)(oa < a) | ((int)(oa == a) & (int)(oi < i));
      a = take ? oa : a;
      i = take ? oi : i;
    }
    bestA[r] = a; bestI[r] = i;
  }

  // Lane 0 writes rows 0..7, lane 16 writes rows 8..15
  if (n16 == 0) {
    const int BL = FCOS_B * FCOS_L;
#pragma unroll
    for (int r = 0; r < 8; ++r) {
      const int li = base + r + 8 * hlf;
      if (li >= FCOS_L) continue;
      const int   g     = bestI[r];
      const float marea = bestA[r];
      const float x = locations[li * 2 + 0];
      const float y = locations[li * 2 + 1];
      float st, lo_, hi_;
      level_of(li, st, lo_, hi_);
      const float inv = 1.0f / st;
      const float bx0 = s_box[g * 4 + 0], by0 = s_box[g * 4 + 1];
      const float bx1 = s_box[g * 4 + 2], by1 = s_box[g * 4 + 3];
      const float l  = (x - bx0) * inv;
      const float t  = (y - by0) * inv;
      const float rr = (bx1 - x) * inv;
      const float bb = (by1 - y) * inv;
      const float lr0 = fmaxf(l, 0.0f),  lr1 = fmaxf(rr, 0.0f);
      const float tb0 = fmaxf(t, 0.0f),  tb1 = fmaxf(bb, 0.0f);
      float c = (fminf(lr0, lr1) / fmaxf(fmaxf(lr0, lr1), 1e-12f)) *
                (fminf(tb0, tb1) / fmaxf(fmaxf(tb0, tb1), 1e-12f));
      c = sqrtf(fmaxf(c, 0.0f));
      c = (c != c) ? 0.0f : c;
      const float lbl = (marea < FCOS_INF) ? (float)s_cls[g] : (float)FCOS_NCLS;
      const int o = b * FCOS_L + li;
      out[o] = lbl;
      out[BL + o * 4 + 0] = l;
      out[BL + o * 4 + 1] = t;
      out[BL + o * 4 + 2] = rr;
      out[BL + o * 4 + 3] = bb;
      out[5 * BL + o] = c;
    }
  }
}

extern "C" void kernel_launch(void* const* d_in, const int* in_sizes, int n_in,
                              void* d_out, int out_size, void* d_ws, size_t ws_size,
                              hipStream_t stream) {
  const float* locations  = (const float*)d_in[0];
  const float* gt_boxes   = (const float*)d_in[1];
  const int*   gt_classes = (const int*)d_in[2];
  float* out = (float*)d_out;

  dim3 grid((FCOS_L + 127) / 128, FCOS_B, 1);  // 8 waves/block * 16 locations/wave
  fcos_assign_wmma<<<grid, 256, 0, stream>>>(locations, gt_boxes, gt_classes, out);
}